// GameNet_4561255269252
// MI455X (gfx1250) — compile-verified
//
#include <hip/hip_runtime.h>

typedef float v2f __attribute__((ext_vector_type(2)));
typedef float v8f __attribute__((ext_vector_type(8)));

#define N_    74
#define V_    54
#define B_    64
#define E_    128
#define O_    64
#define OBS_  14
#define NN_   (N_ * N_)          // 5476
#define NCELL_ (B_ * NN_)        // 350464
#define NE_   (N_ * E_)          // 9472  (adjacency GEMM column count / ld)
#define NPAD_ 80                 // adjacency K padded to 80
#define YROWS_ (NPAD_ * N_)      // 5920 padded y-rows per batch (x128 floats)

// ---------------- utility kernels ----------------

__global__ void zero_f4(float4* __restrict__ p, long n4) {
  long i = (long)blockIdx.x * blockDim.x + threadIdx.x;
  long stride = (long)gridDim.x * blockDim.x;
  float4 z = {0.f, 0.f, 0.f, 0.f};
  for (; i < n4; i += stride) p[i] = z;
}

// cells[r] = b*NN + i[m]*N + j[m],  r = b*count + m
__global__ void build_cells(const int* __restrict__ vi, const int* __restrict__ vj,
                            int count, int* __restrict__ cells) {
  int r = blockIdx.x * blockDim.x + threadIdx.x;
  int total = count * B_;
  if (r >= total) return;
  int b = r / count, m = r - b * count;
  cells[r] = b * NN_ + vi[m] * N_ + vj[m];
}

// yrows[r] = r + (r / NN_) * (YROWS_ - NN_)  : cell -> padded-Y row
__global__ void build_yrows(int* __restrict__ yrows) {
  int r = blockIdx.x * blockDim.x + threadIdx.x;
  if (r >= NCELL_) return;
  yrows[r] = r + (r / NN_) * (YROWS_ - NN_);
}

// zero the 6 pad j-rows per batch of the padded Y buffer (rows 74..79, each NE_ wide)
__global__ void zero_ypad(float* __restrict__ y) {
  int t = blockIdx.x * blockDim.x + threadIdx.x;
  int per_b = (NPAD_ - N_) * NE_;             // 6 * 9472
  if (t >= B_ * per_b) return;
  int b = t / per_b, rem = t - b * per_b;
  y[(long)b * YROWS_ * E_ + (long)N_ * NE_ + rem] = 0.f;
}

// Apad: 80x80, zero padded copy of 74x74 normalized adjacency
__global__ void pad_adj(const float* __restrict__ adj, float* __restrict__ apad) {
  int t = blockIdx.x * blockDim.x + threadIdx.x;
  if (t >= NPAD_ * NPAD_) return;
  int i = t / NPAD_, k = t - i * NPAD_;
  apad[t] = (i < N_ && k < N_) ? adj[i * N_ + k] : 0.f;
}

// pad first-layer weight (kin x 128) to (16 x 128), zero rows beyond kin
__global__ void pad_w1(const float* __restrict__ w, int kin, float* __restrict__ wp) {
  int t = blockIdx.x * blockDim.x + threadIdx.x;
  if (t >= 16 * E_) return;
  int k = t >> 7, n = t & 127;
  wp[t] = (k < kin) ? w[k * E_ + n] : 0.f;
}

// gather obs rows (cin valid features) into dense [rows x 16] zero-padded matrix
__global__ void stage_obs(const float* __restrict__ obs, const int* __restrict__ cells,
                          int rows, int cin, float* __restrict__ staged) {
  int t = blockIdx.x * blockDim.x + threadIdx.x;
  if (t >= rows * 16) return;
  int r = t >> 4, c = t & 15;
  staged[t] = (c < cin) ? obs[(long)cells[r] * OBS_ + c] : 0.f;
}

// ---------------- generic WMMA fp32 GEMM ----------------
// One wave computes a 16x64 output tile (4 n-tiles share one A fragment).
// D = act( gather(A)[M x K] @ W[K x Nc] + bias ), optional output row scatter.
// Requires K % 16 == 0 (true for all configs here: 16, 64, 80, 128).
// No k-clamp: padded operands guarantee all K rows of W are loadable.
// grid.x = M/16 tiles (A-side rows guaranteed loadable), grid.y = Nc/64, grid.z = batches.
__global__ __launch_bounds__(32) void gemm_wmma(
    const float* __restrict__ A, const int* __restrict__ arow, int amul, int lda, long astrideB,
    const float* __restrict__ W, int ldw, long wstrideB,
    const float* __restrict__ bias,
    float* __restrict__ Out, const int* __restrict__ orow, int omul, int ldo, long ostrideB,
    int mstore, int K, int relu)
{
  int lane = threadIdx.x;
  int half = lane >> 4;      // 0: lanes 0-15, 1: lanes 16-31
  int l15  = lane & 15;
  int mt    = blockIdx.x;
  int nbase = blockIdx.y * 64;
  int bz    = blockIdx.z;

  int ra = mt * 16 + l15;    // A-side row this lane reads
  long abase = (arow ? (long)arow[ra] * amul : (long)ra * lda) + (long)bz * astrideB;
  const float* Ap = A + abase;
  // B base: column fixed per lane, rows advance affinely with k (strength-reducible)
  const float* Wb = W + (long)bz * wstrideB + (nbase + l15) + (long)(half * 2) * ldw;

  v8f acc[4];
#pragma unroll
  for (int nt = 0; nt < 4; ++nt) {
    float bv = bias ? bias[nbase + nt * 16 + l15] : 0.f;
#pragma unroll
    for (int v = 0; v < 8; ++v) acc[nt][v] = bv;
  }

  // K is a multiple of 16: process 4 WMMA k-steps (16 k-values) per trip.
  for (int k0 = 0; k0 < K; k0 += 16) {
    v2f av[4];
    v2f bv[4][4];
    const float* wrow = Wb + (long)k0 * ldw;
#pragma unroll
    for (int ks = 0; ks < 4; ++ks) {
      // A fragment: contiguous pair, 8B-aligned for all configs -> b64 load
      av[ks] = *(const v2f*)(Ap + k0 + ks * 4 + half * 2);
#pragma unroll
      for (int nt = 0; nt < 4; ++nt) {
        bv[ks][nt].x = wrow[(long)(ks * 4) * ldw + nt * 16];
        bv[ks][nt].y = wrow[(long)(ks * 4 + 1) * ldw + nt * 16];
      }
    }
#pragma unroll
    for (int ks = 0; ks < 4; ++ks) {
#pragma unroll
      for (int nt = 0; nt < 4; ++nt) {
        // 8 args: (neg_a, A, neg_b, B, c_mod, C, reuse_a, reuse_b)
        acc[nt] = __builtin_amdgcn_wmma_f32_16x16x4_f32(
            false, av[ks], false, bv[ks][nt], (short)0, acc[nt], false, false);
      }
    }
  }

#pragma unroll
  for (int v = 0; v < 8; ++v) {
    int row = mt * 16 + half * 8 + v;   // C/D layout: VGPR v = row v (+8 for high half)
    if (row < mstore) {
      long obase = (orow ? (long)orow[row] * omul : (long)row * ldo) + (long)bz * ostrideB;
#pragma unroll
      for (int nt = 0; nt < 4; ++nt) {
        float val = acc[nt][v];
        if (relu) val = val > 0.f ? val : 0.f;
        Out[obase + nbase + nt * 16 + l15] = val;
      }
    }
  }
}

// ---------------- head combine: out = a + s - mean(a), scattered ----------------
__global__ void combine_heads(const float* __restrict__ a, const float* __restrict__ s,
                              const int* __restrict__ cells, int rows,
                              float* __restrict__ out) {
  int gtid = blockIdx.x * blockDim.x + threadIdx.x;
  int wid = gtid >> 5;          // one wave (32 lanes) per row of 64 channels
  int lane = threadIdx.x & 31;
  if (wid >= rows) return;
  const float* ar = a + (long)wid * O_;
  const float* sr = s + (long)wid * O_;
  float a0 = ar[lane], a1 = ar[lane + 32];
  float s0 = sr[lane], s1 = sr[lane + 32];
  float sum = a0 + a1;
#pragma unroll
  for (int m = 16; m >= 1; m >>= 1) sum += __shfl_xor(sum, m, 32);
  float mean = sum * (1.0f / 64.0f);
  long ob = (long)cells[wid] * O_;
  out[ob + lane]      = a0 + s0 - mean;
  out[ob + lane + 32] = a1 + s1 - mean;
}

// ---------------- launch ----------------

extern "C" void kernel_launch(void* const* d_in, const int* in_sizes, int n_in,
                              void* d_out, int out_size, void* d_ws, size_t ws_size,
                              hipStream_t stream) {
  const float* obs  = (const float*)d_in[0];
  const float* adj  = (const float*)d_in[1];
  const int* node_i = (const int*)d_in[2]; const int* node_j = (const int*)d_in[3];
  const int* edge_i = (const int*)d_in[4]; const int* edge_j = (const int*)d_in[5];
  const int* face_i = (const int*)d_in[6]; const int* face_j = (const int*)d_in[7];
  const int* full_i = (const int*)d_in[8]; const int* full_j = (const int*)d_in[9];
  const float* node_W1 = (const float*)d_in[10]; const float* node_b1 = (const float*)d_in[11];
  const float* node_W2 = (const float*)d_in[12]; const float* node_b2 = (const float*)d_in[13];
  const float* edge_W1 = (const float*)d_in[14]; const float* edge_b1 = (const float*)d_in[15];
  const float* edge_W2 = (const float*)d_in[16]; const float* edge_b2 = (const float*)d_in[17];
  const float* face_W1 = (const float*)d_in[18]; const float* face_b1 = (const float*)d_in[19];
  const float* face_W2 = (const float*)d_in[20]; const float* face_b2 = (const float*)d_in[21];
  const float* act_W1  = (const float*)d_in[22]; const float* act_b1  = (const float*)d_in[23];
  const float* act_W2  = (const float*)d_in[24]; const float* act_b2  = (const float*)d_in[25];
  const float* st_W1   = (const float*)d_in[26]; const float* st_b1   = (const float*)d_in[27];
  const float* st_W2   = (const float*)d_in[28]; const float* st_b2   = (const float*)d_in[29];
  const float* pW      = (const float*)d_in[30]; const float* pb      = (const float*)d_in[31];

  int cn = in_sizes[2], ce = in_sizes[4], cf = in_sizes[6], cu = in_sizes[8];

  // ---- workspace layout ----
  char* ws = (char*)d_ws;
  size_t off = 0;
  auto alloc = [&](size_t bytes) { size_t o = off; off += (bytes + 255) & ~(size_t)255; return o; };
  float* xbuf = (float*)(ws + alloc((size_t)NCELL_ * E_ * 4));          // 171 MB
  char*  ybase = ws + alloc((size_t)B_ * YROWS_ * E_ * 4);              // 194 MB (80 j-rows/batch)
  float* ybuf  = (float*)ybase;
  // sub-buffers alias ybuf (lifetimes: before first / after last use of ybuf)
  float* sgbuf = (float*)ybase;                          // staged obs,  <= cu*B*16  floats
  float* hbuf  = (float*)(ybase + ((size_t)8  << 20));   // hidden, <= cu*B*128 floats
  float* abuf  = (float*)(ybase + ((size_t)40 << 20));   // act head out, cu*B*64
  float* sbuf  = (float*)(ybase + ((size_t)56 << 20));   // st  head out, cu*B*64
  float* apad  = (float*)(ws + alloc(NPAD_ * NPAD_ * 4));
  float* wpadN = (float*)(ws + alloc(16 * E_ * 4));
  float* wpadE = (float*)(ws + alloc(16 * E_ * 4));
  float* wpadF = (float*)(ws + alloc(16 * E_ * 4));
  int* yrows  = (int*)(ws + alloc((size_t)NCELL_ * 4));
  int* cellsN = (int*)(ws + alloc((size_t)cn * B_ * 4));
  int* cellsE = (int*)(ws + alloc((size_t)ce * B_ * 4));
  int* cellsF = (int*)(ws + alloc((size_t)cf * B_ * 4));
  int* cellsU = (int*)(ws + alloc((size_t)cu * B_ * 4));

  // ---- prep ----
  zero_f4<<<4096, 256, 0, stream>>>((float4*)xbuf, (long)NCELL_ * E_ / 4);
  zero_f4<<<2048, 256, 0, stream>>>((float4*)d_out, (long)out_size / 4);
  pad_adj<<<(NPAD_ * NPAD_ + 255) / 256, 256, 0, stream>>>(adj, apad);
  pad_w1<<<8, 256, 0, stream>>>(node_W1, OBS_, wpadN);
  pad_w1<<<8, 256, 0, stream>>>(edge_W1, OBS_, wpadE);
  pad_w1<<<8, 256, 0, stream>>>(face_W1, 6,    wpadF);
  build_yrows<<<(NCELL_ + 255) / 256, 256, 0, stream>>>(yrows);
  build_cells<<<(cn * B_ + 255) / 256, 256, 0, stream>>>(node_i, node_j, cn, cellsN);
  build_cells<<<(ce * B_ + 255) / 256, 256, 0, stream>>>(edge_i, edge_j, ce, cellsE);
  build_cells<<<(cf * B_ + 255) / 256, 256, 0, stream>>>(face_i, face_j, cf, cellsF);
  build_cells<<<(cu * B_ + 255) / 256, 256, 0, stream>>>(full_i, full_j, cu, cellsU);

  // ---- input embed MLPs (scatter into zeroed xbuf) ----
  struct Cls { const int* cells; int count; int cin; const float* wp;
               const float* b1; const float* W2; const float* b2; };
  Cls cls[3] = {
    { cellsN, cn, OBS_, wpadN, node_b1, node_W2, node_b2 },
    { cellsE, ce, OBS_, wpadE, edge_b1, edge_W2, edge_b2 },
    { cellsF, cf, 6,    wpadF, face_b1, face_W2, face_b2 },
  };
  for (int c = 0; c < 3; ++c) {
    int rows = cls[c].count * B_;              // multiple of 16
    stage_obs<<<(rows * 16 + 255) / 256, 256, 0, stream>>>(obs, cls[c].cells, rows,
                                                           cls[c].cin, sgbuf);
    // h = relu(staged[rows x 16] @ W1pad[16 x 128] + b1)
    gemm_wmma<<<dim3(rows / 16, 2, 1), 32, 0, stream>>>(
        sgbuf, nullptr, 0, 16, 0,  cls[c].wp, E_, 0,  cls[c].b1,
        hbuf, nullptr, 0, E_, 0,  rows, 16, 1);
    // x[cells] = relu(h @ W2 + b2)
    gemm_wmma<<<dim3(rows / 16, 2, 1), 32, 0, stream>>>(
        hbuf, nullptr, 0, E_, 0,  cls[c].W2, E_, 0,  cls[c].b2,
        xbuf, cls[c].cells, E_, 0, 0,  rows, E_, 1);
  }

  // pad j-rows of Y must be zero before propagation (embed aliased this region)
  zero_ypad<<<(B_ * (NPAD_ - N_) * NE_ + 255) / 256, 256, 0, stream>>>(ybuf);

  // ---- 4 propagation layers ----
  for (int l = 0; l < 4; ++l) {
    // y[padded rows] = x @ pW[l] + pb[l]  over all 350464 cells (zero cells still get pb!)
    gemm_wmma<<<dim3(NCELL_ / 16, 2, 1), 32, 0, stream>>>(
        xbuf, nullptr, 0, E_, 0,
        pW + (size_t)l * E_ * E_, E_, 0,  pb + (size_t)l * E_,
        ybuf, yrows, E_, 0, 0,  NCELL_, E_, 0);
    // x[b,i,:,:] = relu( Apad[80x80] @ ypad_b[80 x 9472] )   per batch (grid.z)
    gemm_wmma<<<dim3(5, NE_ / 64, B_), 32, 0, stream>>>(
        apad, nullptr, 0, NPAD_, 0,
        ybuf, NE_, (long)YROWS_ * E_,  nullptr,
        xbuf, nullptr, 0, NE_, (long)NN_ * E_,  N_, NPAD_, 1);
  }

  // ---- heads on gathered full positions ----
  int rows = cu * B_;
  // act head
  gemm_wmma<<<dim3(rows / 16, 1, 1), 32, 0, stream>>>(
      xbuf, cellsU, E_, 0, 0,  act_W1, O_, 0,  act_b1,
      hbuf, nullptr, 0, O_, 0,  rows, E_, 1);
  gemm_wmma<<<dim3(rows / 16, 1, 1), 32, 0, stream>>>(
      hbuf, nullptr, 0, O_, 0,  act_W2, O_, 0,  act_b2,
      abuf, nullptr, 0, O_, 0,  rows, O_, 0);
  // state head
  gemm_wmma<<<dim3(rows / 16, 1, 1), 32, 0, stream>>>(
      xbuf, cellsU, E_, 0, 0,  st_W1, O_, 0,  st_b1,
      hbuf, nullptr, 0, O_, 0,  rows, E_, 1);
  gemm_wmma<<<dim3(rows / 16, 1, 1), 32, 0, stream>>>(
      hbuf, nullptr, 0, O_, 0,  st_W2, O_, 0,  st_b2,
      sbuf, nullptr, 0, O_, 0,  rows, O_, 0);

  // out = A + S - mean(A) at full positions (d_out pre-zeroed elsewhere)
  combine_heads<<<(rows * 32 + 255) / 256, 256, 0, stream>>>(abuf, sbuf, cellsU, rows,
                                                             (float*)d_out);
}